// BoundaryChunker_79199196938420
// MI455X (gfx1250) — compile-verified
//
#include <hip/hip_runtime.h>
#include <hip/hip_bf16.h>
#include <stdint.h>

#define B_SZ  8
#define L_SZ  4096
#define D_SZ  1024
#define POOLW 5
#define EPSV  1e-6f
#define BK    32
#define APITCH 36   // LDS row pitch (floats): mult of 4 (b128-aligned), conflict-free

typedef float v2f __attribute__((ext_vector_type(2)));
typedef float v8f __attribute__((ext_vector_type(8)));
typedef int   v4i_gcc __attribute__((vector_size(16)));   // matches builtin pointee

#if defined(__has_builtin)
#if __has_builtin(__builtin_amdgcn_global_load_async_to_lds_b128) && \
    __has_builtin(__builtin_amdgcn_s_wait_asynccnt)
#define USE_ASYNC_LDS 1
#endif
#endif

// Stage 16B global -> LDS. Async DMA path on CDNA5 (ASYNCcnt), sync fallback.
__device__ __forceinline__ void stage16(const float* __restrict__ gsrc, float* ldst) {
#ifdef USE_ASYNC_LDS
    __builtin_amdgcn_global_load_async_to_lds_b128(
        (__attribute__((address_space(1))) v4i_gcc*)gsrc,   // C-style cast drops const
        (__attribute__((address_space(3))) v4i_gcc*)ldst,   // generic->AS3 addrspacecast
        /*offset=*/0, /*cpol=*/0);
#else
    *(float4*)ldst = *(const float4*)gsrc;
#endif
}

__device__ __forceinline__ void wait_stage() {
#ifdef USE_ASYNC_LDS
    __builtin_amdgcn_s_wait_asynccnt(0);
#endif
}

// ---------------------------------------------------------------------------
// Kernel A: per-batch boundary compaction. One wave32 per batch.
// ---------------------------------------------------------------------------
__global__ __launch_bounds__(32)
void compact_kernel(const uint8_t* __restrict__ bmask,
                    const float*   __restrict__ conf_in,
                    int*           __restrict__ counts,
                    int*           __restrict__ pos,
                    float*         __restrict__ mask_out,
                    float*         __restrict__ idx_out,
                    float*         __restrict__ conf_out,
                    int U) {
    const int b    = blockIdx.x;
    const int lane = threadIdx.x;                    // 0..31
    const uint8_t* m = bmask + (size_t)b * L_SZ;

    int base = 0;                                    // wave-uniform running count
    for (int t0 = 0; t0 < L_SZ; t0 += 32) {
        const int  t   = t0 + lane;
        const bool hit = (m[t] != 0);
        const unsigned bal  = __builtin_amdgcn_ballot_w32(hit);
        const int      rank = __popc(bal & ((1u << lane) - 1u));
        if (hit) {
            const int slot = base + rank;
            if (slot < U) {
                pos[b * U + slot]      = t;
                idx_out[b * U + slot]  = (float)t;
                conf_out[b * U + slot] = conf_in[(size_t)b * L_SZ + t];
            }
        }
        base += __popc(bal);
    }
    if (lane == 0) counts[b] = base;

    for (int u = lane; u < U; u += 32) {
        const bool valid = (u < base);
        mask_out[b * U + u] = valid ? 1.0f : 0.0f;
        if (!valid) {
            idx_out[b * U + u]  = 0.0f;
            conf_out[b * U + u] = 0.0f;
        }
    }
}

// ---------------------------------------------------------------------------
__global__ void fill_zero(float* __restrict__ p, int n) {
    const int i = blockIdx.x * blockDim.x + threadIdx.x;
    if (i < n) p[i] = 0.0f;
}

// ---------------------------------------------------------------------------
// Kernel B: windowed weighted pooling ONLY at boundary slots.
// ---------------------------------------------------------------------------
__global__ __launch_bounds__(256)
void pool_kernel(const float* __restrict__ emb,
                 const float* __restrict__ cs,
                 const int*   __restrict__ counts,
                 const int*   __restrict__ pos,
                 float*       __restrict__ pooled,
                 int U) {
    const int u = blockIdx.x;
    const int b = blockIdx.y;
    if (u >= counts[b]) return;                      // row pre-zeroed

    const int n     = b * U + u;
    const int t     = pos[n];
    const int start = (t - (POOLW - 1) > 0) ? (t - (POOLW - 1)) : 0;

    float wsum = 0.0f;
    for (int s = start; s <= t; ++s) wsum += cs[b * L_SZ + s];
    const float inv = 1.0f / (wsum + EPSV);

    const int d0 = threadIdx.x * 4;
    float4 acc = make_float4(0.f, 0.f, 0.f, 0.f);
    for (int s = start; s <= t; ++s) {
        const float  w = cs[b * L_SZ + s];
        const float4 e = *(const float4*)(emb + ((size_t)b * L_SZ + s) * D_SZ + d0);
        acc.x += w * e.x; acc.y += w * e.y; acc.z += w * e.z; acc.w += w * e.w;
    }
    acc.x *= inv; acc.y *= inv; acc.z *= inv; acc.w *= inv;
    *(float4*)(pooled + (size_t)n * D_SZ + d0) = acc;
}

// ---------------------------------------------------------------------------
// Kernel C: chunks = pooled @ W^T + bias, masked by slot validity.
// Double-buffered LDS A-tiles filled by GLOBAL_LOAD_ASYNC_TO_LDS_B128;
// fp32 WMMA (V_WMMA_F32_16X16X4_F32) for the inner product.
// ---------------------------------------------------------------------------
__global__ __launch_bounds__(128)
void gemm_kernel(const float* __restrict__ A,      // pooled, Mpad x D
                 const float* __restrict__ W,      // D x D row-major (j, k)
                 const float* __restrict__ bias,   // D
                 const int*   __restrict__ counts, // B
                 float*       __restrict__ out,    // Nrows x D
                 int U, int Nrows) {
    __shared__ float As[2][16][APITCH];

    const int tid  = threadIdx.x;
    const int wave = tid >> 5;
    const int lane = tid & 31;
    const int half = lane >> 4;                    // 0 | 1
    const int l16  = lane & 15;

    const int m0   = blockIdx.y * 16;
    const int jcol = blockIdx.x * 64 + wave * 16 + l16;

    const float* __restrict__ wrow = W + (size_t)jcol * D_SZ;

    // staging coordinates: 128 threads x one 16B transfer = 16 x 32 tile
    const int srow = tid >> 3;
    const int sc4  = (tid & 7) * 4;
    const float* __restrict__ aRow = A + (size_t)(m0 + srow) * D_SZ + sc4;

    v8f acc = {};

    // prologue: stage chunk 0 into buffer 0
    stage16(aRow, &As[0][srow][sc4]);
    wait_stage();
    __syncthreads();

    for (int k0 = 0; k0 < D_SZ; k0 += BK) {
        const int buf = (k0 / BK) & 1;

        // kick off DMA of the next A chunk into the other buffer
        if (k0 + BK < D_SZ)
            stage16(aRow + k0 + BK, &As[buf ^ 1][srow][sc4]);

        // prefetch next W chunk (global_prefetch_b8)
        if (k0 + BK < D_SZ) __builtin_prefetch(wrow + k0 + BK, 0, 1);

#pragma unroll
        for (int kk = 0; kk < BK; kk += 4) {
            const int kl = kk + half * 2;
            v2f a;
            a.x = As[buf][l16][kl];
            a.y = As[buf][l16][kl + 1];
            const v2f bb = *(const v2f*)(wrow + k0 + kl);
            acc = __builtin_amdgcn_wmma_f32_16x16x4_f32(
                    /*neg_a=*/false, a, /*neg_b=*/false, bb,
                    /*c_mod=*/(short)0, acc,
                    /*reuse_a=*/false, /*reuse_b=*/false);
        }

        wait_stage();      // our next-buffer DMA landed
        __syncthreads();   // everyone done reading buf / writing buf^1
    }

    // epilogue: add bias, mask invalid slots (u >= counts[b]) to exactly 0
    const float bj = bias[jcol];
#pragma unroll
    for (int r = 0; r < 8; ++r) {
        const int m = m0 + r + half * 8;
        if (m < Nrows) {
            const int  u     = m % U;
            const int  bidx  = m / U;
            const bool valid = (u < counts[bidx]);
            out[(size_t)m * D_SZ + jcol] = valid ? (acc[r] + bj) : 0.0f;
        }
    }
}

// ---------------------------------------------------------------------------
extern "C" void kernel_launch(void* const* d_in, const int* in_sizes, int n_in,
                              void* d_out, int out_size, void* d_ws, size_t ws_size,
                              hipStream_t stream) {
    const float*   emb   = (const float*)d_in[0];
    const uint8_t* bmask = (const uint8_t*)d_in[1];   // jax bool -> 1 byte/elem
    const float*   cs    = (const float*)d_in[2];
    const float*   conf  = (const float*)d_in[3];
    const float*   W     = (const float*)d_in[4];
    const float*   bias  = (const float*)d_in[5];
    float*         out   = (float*)d_out;

    // out_size = B*U*D + 3*B*U  ->  U
    const int U     = out_size / (B_SZ * (D_SZ + 3));
    const int Nrows = B_SZ * U;
    const int Mpad  = (Nrows + 15) & ~15;

    // d_out layout: chunks | slot_mask | idx | conf  (all f32)
    float* chunks   = out;
    float* mask_out = out + (size_t)Nrows * D_SZ;
    float* idx_out  = mask_out + Nrows;
    float* conf_out = idx_out + Nrows;

    // workspace: counts | pos | pooled (256B aligned segments)
    uint8_t* ws      = (uint8_t*)d_ws;
    int*     counts  = (int*)ws;
    int*     pos     = (int*)(ws + 256);
    size_t   posB    = (((size_t)Nrows * sizeof(int)) + 255) & ~(size_t)255;
    float*   pooled  = (float*)(ws + 256 + posB);

    compact_kernel<<<B_SZ, 32, 0, stream>>>(bmask, conf, counts, pos,
                                            mask_out, idx_out, conf_out, U);

    const int nfill = Mpad * D_SZ;
    fill_zero<<<(nfill + 255) / 256, 256, 0, stream>>>(pooled, nfill);

    pool_kernel<<<dim3(U, B_SZ), 256, 0, stream>>>(emb, cs, counts, pos, pooled, U);

    gemm_kernel<<<dim3(D_SZ / 64, Mpad / 16), 128, 0, stream>>>(
        pooled, W, bias, counts, chunks, U, Nrows);
}